// CausalSelfAttention_46806553592222
// MI455X (gfx1250) — compile-verified
//
#include <hip/hip_runtime.h>
#include <math.h>
#include <stdint.h>

// ---------------------------------------------------------------- constants
#define N_EMBD 1024
#define N_HEAD 16
#define HEADD  64
#define BATCH  4
#define SEQ    2048
#define C3     (3 * N_EMBD)

typedef __attribute__((ext_vector_type(2)))  float        v2f;
typedef __attribute__((ext_vector_type(8)))  float        v8f;
typedef __attribute__((ext_vector_type(4)))  unsigned int u32x4;
typedef __attribute__((ext_vector_type(8)))  int          i32x8;
typedef __attribute__((ext_vector_type(4)))  int          i32x4;

#define WMMA_F32X4(a, b, c) \
    __builtin_amdgcn_wmma_f32_16x16x4_f32(false, (a), false, (b), (short)0, (c), false, false)

static __device__ inline void wave_lds_fence() {
    asm volatile("s_wait_dscnt 0x0" ::: "memory");
}

#if defined(__AMDGCN__) && __has_builtin(__builtin_amdgcn_tensor_load_to_lds)
#define HAVE_TDM 1
#else
#define HAVE_TDM 0
#endif

// ---------------------------------------------------------------- GEMM
// C[M,N] = A[M,K]*B[K,N] + bias[N], fp32 row-major.
// 256 thr = 8 waves in 2(M) x 4(N); wave tile 64x64 (4x4 of 16x16 WMMA).
// Macro-tile 128(M) x 256(N), K-tile 32, double-buffered LDS fed by TDM.
#define KT       32
#define MACM     128
#define MACN     256
#define ASTRIDE  33                 // A LDS row stride (TDM pad: +1 DW / 32 DW)
#define BSTRIDE  264                // B LDS row stride (TDM pad: +8 DW / 256 DW)
#define A_FLOATS (MACM * ASTRIDE)   // 4224
#define B_FLOATS (KT * BSTRIDE)     // 8448
#define GEMM_LDS_BYTES ((2 * A_FLOATS + 2 * B_FLOATS) * 4)  // 101376

#if HAVE_TDM
// Build a 2D tensor-DMA descriptor and issue TENSOR_LOAD_TO_LDS (6-arg form).
// tile_w contiguous elements per row, tile_h rows, row stride in elements.
static __device__ inline void tdm_load_2d(unsigned lds_off, const float* gptr,
                                          unsigned tile_w, unsigned tile_h,
                                          unsigned long long row_stride_el,
                                          unsigned pad_int_enc, unsigned pad_amt_enc,
                                          bool pad_en)
{
    unsigned long long ga = (unsigned long long)(uintptr_t)gptr;
    u32x4 g0;
    g0[0] = 1u;                                        // count=1, user desc
    g0[1] = lds_off;                                   // lds_addr (bytes)
    g0[2] = (unsigned)(ga & 0xFFFFFFFFu);              // global_addr[31:0]
    g0[3] = (unsigned)((ga >> 32) & 0x1FFFFFFu) | (2u << 30);  // [56:32] | type=2
    unsigned w0 = (2u << 16);                          // data_size = 4B
    if (pad_en) w0 |= (1u << 20) | (pad_int_enc << 22) | (pad_amt_enc << 25);
    const unsigned td0 = 0x7FFFFFFFu, td1 = 0x7FFFFFFFu;  // huge dims: no OOB clip
    i32x8 g1;
    g1[0] = (int)w0;                                   // mask=0, data_size, pad
    g1[1] = (int)((td0 & 0xFFFFu) << 16);              // bar_addr=0, dim0[15:0]
    g1[2] = (int)((td0 >> 16) | ((td1 & 0xFFFFu) << 16));
    g1[3] = (int)((td1 >> 16) | (tile_w << 16));       // dim1[31:16], tile_dim0
    g1[4] = (int)tile_h;                               // tile_dim1, tile_dim2=0
    g1[5] = (int)(unsigned)(row_stride_el & 0xFFFFFFFFu);      // dim0_stride lo
    g1[6] = (int)(unsigned)((row_stride_el >> 32) & 0xFFFFu);  // dim0_stride hi
    g1[7] = 0;
    i32x4 z4 = {0, 0, 0, 0};
    i32x8 z8 = {0, 0, 0, 0, 0, 0, 0, 0};
    __builtin_amdgcn_tensor_load_to_lds(g0, g1, z4, z4, z8, 0);
}
#endif

__global__ __launch_bounds__(256) void gemm_bias_f32(
    const float* __restrict__ A, const float* __restrict__ Bm,
    const float* __restrict__ bias, float* __restrict__ Cm,
    int M, int N, int K)
{
    extern __shared__ float lds[];

    const int tid   = threadIdx.x;
    const int wave  = tid >> 5;
    const int lane  = tid & 31;
    const int waveM = wave >> 2;       // 0..1
    const int waveN = wave & 3;        // 0..3
    const int hl    = lane >> 4;
    const int l15   = lane & 15;
    const int kb    = hl * 2;

    const int blockN = blockIdx.x * MACN;
    const int blockM = blockIdx.y * MACM;

    v8f acc[4][4];
    #pragma unroll
    for (int i = 0; i < 4; ++i)
        #pragma unroll
        for (int j = 0; j < 4; ++j)
            acc[i][j] = (v8f){0.f,0.f,0.f,0.f,0.f,0.f,0.f,0.f};

#if HAVE_TDM
    // prologue: wave 0 kicks DMA for k-block 0 into buffer 0
    if (wave == 0) {
        tdm_load_2d(0,                A + (size_t)blockM * K, KT,   MACM, (unsigned long long)K, 4, 0, true);
        tdm_load_2d(2 * A_FLOATS * 4, Bm + blockN,            MACN, KT,   (unsigned long long)N, 7, 7, true);
    }
#endif

    for (int k0 = 0; k0 < K; k0 += KT) {
        const int buf = (k0 / KT) & 1;
        float* __restrict__ As = lds + buf * A_FLOATS;
        float* __restrict__ Bs = lds + 2 * A_FLOATS + buf * B_FLOATS;

#if HAVE_TDM
        if (wave == 0) {
            const int kn = k0 + KT;
            if (kn < K) {
                const int nb = buf ^ 1;
                tdm_load_2d(nb * A_FLOATS * 4,                  A + (size_t)blockM * K + kn,  KT,   MACM, (unsigned long long)K, 4, 0, true);
                tdm_load_2d((2 * A_FLOATS + nb * B_FLOATS) * 4, Bm + (size_t)kn * N + blockN, MACN, KT,   (unsigned long long)N, 7, 7, true);
                __builtin_amdgcn_s_wait_tensorcnt(2);  // current pair done
            } else {
                __builtin_amdgcn_s_wait_tensorcnt(0);
            }
        }
        __syncthreads();
#else
        __syncthreads();   // WAR vs previous compute
        #pragma unroll
        for (int j = 0; j < 4; ++j) {            // A: 128x32 = 1024 float4
            int idx4 = tid + 256 * j;
            int row  = idx4 >> 3, c4 = idx4 & 7;
            float4 v = *(const float4*)(A + (size_t)(blockM + row) * K + k0 + c4 * 4);
            float* d = &As[row * ASTRIDE + c4 * 4];
            d[0] = v.x; d[1] = v.y; d[2] = v.z; d[3] = v.w;
        }
        #pragma unroll
        for (int j = 0; j < 8; ++j) {            // B: 32x256 = 2048 float4
            int idx4 = tid + 256 * j;
            int krow = idx4 >> 6, c4 = idx4 & 63;
            float4 v = *(const float4*)(Bm + (size_t)(k0 + krow) * N + blockN + c4 * 4);
            float* d = &Bs[krow * BSTRIDE + c4 * 4];
            d[0] = v.x; d[1] = v.y; d[2] = v.z; d[3] = v.w;
        }
        __syncthreads();
#endif

        #pragma unroll
        for (int ks = 0; ks < KT; ks += 4) {
            v2f af[4], bf[4];
            #pragma unroll
            for (int i = 0; i < 4; ++i) {
                const float* p = &As[(waveM * 64 + i * 16 + l15) * ASTRIDE + ks + kb];
                af[i][0] = p[0]; af[i][1] = p[1];
            }
            #pragma unroll
            for (int j = 0; j < 4; ++j) {
                const float* p = &Bs[(ks + kb) * BSTRIDE + waveN * 64 + j * 16 + l15];
                bf[j][0] = p[0]; bf[j][1] = p[BSTRIDE];
            }
            #pragma unroll
            for (int i = 0; i < 4; ++i)
                #pragma unroll
                for (int j = 0; j < 4; ++j)
                    acc[i][j] = WMMA_F32X4(af[i], bf[j], acc[i][j]);
        }
        __syncthreads();   // compute done before next DMA overwrites this buffer
    }

    #pragma unroll
    for (int i = 0; i < 4; ++i)
        #pragma unroll
        for (int j = 0; j < 4; ++j) {
            int col  = blockN + waveN * 64 + j * 16 + l15;
            float bv = bias[col];
            #pragma unroll
            for (int r = 0; r < 8; ++r) {
                int row = blockM + waveM * 64 + i * 16 + r + hl * 8;
                Cm[(size_t)row * N + col] = acc[i][j][r] + bv;
            }
        }
}

// ---------------------------------------------------------------- attention
__global__ __launch_bounds__(128) void attn_flash_f32(
    const float* __restrict__ qkv, float* __restrict__ out)
{
    const int tid  = threadIdx.x;
    const int wave = tid >> 5;
    const int lane = tid & 31;
    const int hl   = lane >> 4;
    const int l15  = lane & 15;
    const int kb   = hl * 2;

    const int blocksPerBH = (SEQ / 16) / 4;    // 32
    const int bh    = blockIdx.x / blocksPerBH;
    const int qtile = (blockIdx.x % blocksPerBH) * 4 + wave;
    const int b = bh / N_HEAD, h = bh % N_HEAD;
    const int q0 = qtile * 16;

    const size_t rs = (size_t)C3;
    const float* qB = qkv + (size_t)b * SEQ * rs + h * HEADD;
    const float* kB = qB + N_EMBD;
    const float* vB = qB + 2 * N_EMBD;

    v2f qf[16];
    {
        const float* qr = qB + (size_t)(q0 + l15) * rs;
        const float sc = 0.125f;
        #pragma unroll
        for (int s = 0; s < 16; ++s) {
            qf[s][0] = qr[4 * s + kb + 0] * sc;
            qf[s][1] = qr[4 * s + kb + 1] * sc;
        }
    }

    v8f o[4];
    #pragma unroll
    for (int c = 0; c < 4; ++c) o[c] = (v8f){0.f,0.f,0.f,0.f,0.f,0.f,0.f,0.f};
    float m_[8], l_[8];
    #pragma unroll
    for (int r = 0; r < 8; ++r) { m_[r] = -__builtin_inff(); l_[r] = 0.f; }

    __shared__ float Plds[4][16 * 17];
    float* pl = &Plds[wave][0];

    for (int kt = 0; kt <= qtile; ++kt) {
        const int k0 = kt * 16;

        v8f s = (v8f){0.f,0.f,0.f,0.f,0.f,0.f,0.f,0.f};
        {
            const float* kr = kB + (size_t)(k0 + l15) * rs;
            #pragma unroll
            for (int st = 0; st < 16; ++st) {
                v2f kf; kf[0] = kr[4 * st + kb + 0]; kf[1] = kr[4 * st + kb + 1];
                s = WMMA_F32X4(qf[st], kf, s);
            }
        }

        if (kt == qtile) {
            #pragma unroll
            for (int r = 0; r < 8; ++r) {
                int row_ = r + hl * 8;
                s[r] = (l15 <= row_) ? s[r] : -__builtin_inff();
            }
        }

        float corr[8];
        #pragma unroll
        for (int r = 0; r < 8; ++r) {
            float v = s[r];
            v = fmaxf(v, __shfl_xor(v, 1, 32));
            v = fmaxf(v, __shfl_xor(v, 2, 32));
            v = fmaxf(v, __shfl_xor(v, 4, 32));
            v = fmaxf(v, __shfl_xor(v, 8, 32));
            float mn = fmaxf(m_[r], v);
            corr[r]  = __expf(m_[r] - mn);
            m_[r]    = mn;
            s[r]     = __expf(s[r] - mn);
        }
        #pragma unroll
        for (int r = 0; r < 8; ++r) {
            float v = s[r];
            v += __shfl_xor(v, 1, 32);
            v += __shfl_xor(v, 2, 32);
            v += __shfl_xor(v, 4, 32);
            v += __shfl_xor(v, 8, 32);
            l_[r] = l_[r] * corr[r] + v;
        }
        #pragma unroll
        for (int c = 0; c < 4; ++c)
            #pragma unroll
            for (int r = 0; r < 8; ++r) o[c][r] *= corr[r];

        #pragma unroll
        for (int r = 0; r < 8; ++r) pl[(r + hl * 8) * 17 + l15] = s[r];
        wave_lds_fence();

        #pragma unroll
        for (int s4 = 0; s4 < 4; ++s4) {
            v2f pf;
            pf[0] = pl[l15 * 17 + 4 * s4 + kb + 0];
            pf[1] = pl[l15 * 17 + 4 * s4 + kb + 1];
            const float* vr0 = vB + (size_t)(k0 + 4 * s4 + kb) * rs;
            const float* vr1 = vr0 + rs;
            #pragma unroll
            for (int c = 0; c < 4; ++c) {
                v2f vf; vf[0] = vr0[c * 16 + l15]; vf[1] = vr1[c * 16 + l15];
                o[c] = WMMA_F32X4(pf, vf, o[c]);
            }
        }
        wave_lds_fence();
    }

    float* ob = out + (size_t)b * SEQ * N_EMBD + h * HEADD;
    #pragma unroll
    for (int c = 0; c < 4; ++c)
        #pragma unroll
        for (int r = 0; r < 8; ++r) {
            int row = q0 + r + hl * 8;
            ob[(size_t)row * N_EMBD + c * 16 + l15] = o[c][r] / l_[r];
        }
}

// ---------------------------------------------------------------- launch
extern "C" void kernel_launch(void* const* d_in, const int* in_sizes, int n_in,
                              void* d_out, int out_size, void* d_ws, size_t ws_size,
                              hipStream_t stream) {
    const float* x      = (const float*)d_in[0];
    const float* w_attn = (const float*)d_in[1];
    const float* b_attn = (const float*)d_in[2];
    const float* w_proj = (const float*)d_in[3];
    const float* b_proj = (const float*)d_in[4];
    float* out = (float*)d_out;

    const int M = BATCH * SEQ;                               // 8192
    float* qkv      = (float*)d_ws;                          // [B,T,3C]
    float* attn_out = qkv + (size_t)BATCH * SEQ * C3;        // [B,T,C]

    // idempotent, deterministic, not a stream op (capture-safe)
    (void)hipFuncSetAttribute((const void*)gemm_bias_f32,
                              hipFuncAttributeMaxDynamicSharedMemorySize,
                              GEMM_LDS_BYTES);

    gemm_bias_f32<<<dim3(C3 / MACN, M / MACM), 256, GEMM_LDS_BYTES, stream>>>(
        x, w_attn, b_attn, qkv, M, C3, N_EMBD);

    attn_flash_f32<<<dim3(BATCH * N_HEAD * (SEQ / 64)), 128, 0, stream>>>(
        qkv, attn_out);

    gemm_bias_f32<<<dim3(N_EMBD / MACN, M / MACM), 256, GEMM_LDS_BYTES, stream>>>(
        attn_out, w_proj, b_proj, out, M, N_EMBD, N_EMBD);
}